// Block_40209483825240
// MI455X (gfx1250) — compile-verified
//
#include <hip/hip_runtime.h>
#include <hip/hip_bf16.h>
#include <math.h>

// MI455X / gfx1250, wave32. All matrix math on V_WMMA_F32_16X16X4_F32 (native fp32 WMMA).

typedef __attribute__((ext_vector_type(2))) float v2f;
typedef __attribute__((ext_vector_type(8))) float v8f;

#define BATCH   2
#define TSEQ    2048
#define DMODEL  1024
#define NHEADS  16
#define NKV     4
#define HEADD   64

__device__ __forceinline__ v8f wmma_f32(v2f a, v2f b, v8f c) {
  // 8 args: (neg_a, A, neg_b, B, c_mod, C, reuse_a, reuse_b)
  return __builtin_amdgcn_wmma_f32_16x16x4_f32(false, a, false, b, (short)0, c, false, false);
}

// ---------------------------------------------------------------------------
// RoPE: in (B,T,nheads,64) -> out (B,nheads,T,64), rotated.
// sin/cos table: freq index i = d & 31, theta = 10000^(i/32), ang = t/theta.
// ---------------------------------------------------------------------------
__global__ void rope_kernel(const float* __restrict__ in, float* __restrict__ out,
                            int nheads) {
  int idx = blockIdx.x * blockDim.x + threadIdx.x;
  int total = BATCH * TSEQ * nheads * HEADD;
  if (idx >= total) return;
  int d = idx & 63;
  int t = (idx >> 6) & (TSEQ - 1);
  int h = (idx / (64 * TSEQ)) % nheads;
  int b = idx / (64 * TSEQ * nheads);
  const float* src = in + (((size_t)(b * TSEQ + t) * nheads + h) << 6);
  int i = d & 31;
  float freq = powf(10000.0f, -(float)i * (1.0f / 32.0f));
  float ang = (float)t * freq;
  float s = sinf(ang);
  float c = cosf(ang);
  float x = src[d];
  float xr = (d < 32) ? -src[d + 32] : src[d - 32];
  out[(((size_t)(b * nheads + h) * TSEQ + t) << 6) + d] = x * c + xr * s;
}

// ---------------------------------------------------------------------------
// Flash-style causal GQA attention. One wave (32 threads) per
// (batch, head, 16-row query tile). fp32 WMMA throughout.
//   qr: (B, NHEADS, T, 64) rotated, kr: (B, NKV, T, 64) rotated,
//   v_in: (B, T, NKV, 64) raw, xo: (B, T, DMODEL) attention output.
// ---------------------------------------------------------------------------
__global__ void __launch_bounds__(32) attn_kernel(const float* __restrict__ qr,
                                                  const float* __restrict__ kr,
                                                  const float* __restrict__ v_in,
                                                  float* __restrict__ xo) {
  __shared__ float pt[16][17];  // P exchange tile (C/D layout -> A layout), padded
  const int lane  = threadIdx.x;
  const int tile  = blockIdx.x & (TSEQ / 16 - 1);
  const int h     = (blockIdx.x >> 7) & (NHEADS - 1);
  const int b     = blockIdx.x >> 11;
  const int kvh   = h >> 2;              // GQA: 4 Q heads per KV head
  const int qbase = tile << 4;
  const int lm    = lane & 15;           // A:M / B:N / C:N index
  const int kofs  = (lane >> 4) << 1;    // K offset within a K=4 chunk
  const int mhi   = (lane >> 4) << 3;    // C/D: M = r + mhi

  // Q as 16 A-fragments (K=64 in chunks of 4), pre-scaled by 1/sqrt(64)
  v2f qa[16];
  {
    const float* Qp = qr + (((size_t)(b * NHEADS + h) * TSEQ + qbase + lm) << 6);
#pragma unroll
    for (int c = 0; c < 16; ++c) {
      qa[c][0] = Qp[4 * c + kofs + 0] * 0.125f;
      qa[c][1] = Qp[4 * c + kofs + 1] * 0.125f;
    }
  }

  v8f o0 = {}, o1 = {}, o2 = {}, o3 = {};
  float mrow[8], lrow[8];
#pragma unroll
  for (int r = 0; r < 8; ++r) { mrow[r] = -3.0e38f; lrow[r] = 0.0f; }

  for (int kt = 0; kt <= tile; ++kt) {
    const int kvbase = kt << 4;

    // S = Q K^T : B-fragment b[j] = K[n][k], n = lm, k = 4c + kofs + j
    const float* Kp = kr + (((size_t)(b * NKV + kvh) * TSEQ + kvbase + lm) << 6);
    v8f acc = {};
#pragma unroll
    for (int c = 0; c < 16; ++c) {
      v2f kb;
      kb[0] = Kp[4 * c + kofs + 0];
      kb[1] = Kp[4 * c + kofs + 1];
      acc = wmma_f32(qa[c], kb, acc);
    }

    // causal mask + online softmax (rows live in 16-lane half-wave groups)
    float p[8];
#pragma unroll
    for (int r = 0; r < 8; ++r) {
      int m = r + mhi;
      float sv = acc[r];
      if (kvbase + lm > qbase + m) sv = -1.0e9f;
      float mx = sv;
      mx = fmaxf(mx, __shfl_xor(mx, 1, 32));
      mx = fmaxf(mx, __shfl_xor(mx, 2, 32));
      mx = fmaxf(mx, __shfl_xor(mx, 4, 32));
      mx = fmaxf(mx, __shfl_xor(mx, 8, 32));
      float newm = fmaxf(mrow[r], mx);
      float pv = __expf(sv - newm);
      float rs = pv;
      rs += __shfl_xor(rs, 1, 32);
      rs += __shfl_xor(rs, 2, 32);
      rs += __shfl_xor(rs, 4, 32);
      rs += __shfl_xor(rs, 8, 32);
      float corr = __expf(mrow[r] - newm);
      lrow[r] = lrow[r] * corr + rs;
      mrow[r] = newm;
      o0[r] *= corr; o1[r] *= corr; o2[r] *= corr; o3[r] *= corr;
      p[r] = pv;
    }

    // C/D layout -> row-major 16x16 in LDS, reload as A-fragments
#pragma unroll
    for (int r = 0; r < 8; ++r) pt[r + mhi][lm] = p[r];
    __syncthreads();  // single wave: DS ordering (s_wait_dscnt), barrier ~ nop

    // O += P @ V ; V B-fragment b[j] = V[k][d], k = kvbase+4c+kofs+j, d = nt*16+lm
#pragma unroll
    for (int c = 0; c < 4; ++c) {
      v2f pa;
      pa[0] = pt[lm][4 * c + kofs + 0];
      pa[1] = pt[lm][4 * c + kofs + 1];
      int k0 = kvbase + 4 * c + kofs;
      const float* V0 = v_in + (((size_t)(b * TSEQ + k0 + 0) * NKV + kvh) << 6);
      const float* V1 = v_in + (((size_t)(b * TSEQ + k0 + 1) * NKV + kvh) << 6);
      {
        v2f vb; vb[0] = V0[ 0 + lm]; vb[1] = V1[ 0 + lm]; o0 = wmma_f32(pa, vb, o0);
      }
      { v2f vb; vb[0] = V0[16 + lm]; vb[1] = V1[16 + lm]; o1 = wmma_f32(pa, vb, o1); }
      { v2f vb; vb[0] = V0[32 + lm]; vb[1] = V1[32 + lm]; o2 = wmma_f32(pa, vb, o2); }
      { v2f vb; vb[0] = V0[48 + lm]; vb[1] = V1[48 + lm]; o3 = wmma_f32(pa, vb, o3); }
    }
    __syncthreads();
  }

  // normalize and store into (B,T,DMODEL)
#pragma unroll
  for (int r = 0; r < 8; ++r) {
    int m = r + mhi;
    float inv = 1.0f / lrow[r];
    size_t row = (size_t)(b * TSEQ + qbase + m) * DMODEL + h * HEADD;
    xo[row +  0 + lm] = o0[r] * inv;
    xo[row + 16 + lm] = o1[r] * inv;
    xo[row + 32 + lm] = o2[r] * inv;
    xo[row + 48 + lm] = o3[r] * inv;
  }
}

// ---------------------------------------------------------------------------
// Output projection: Y(4096x1024) = X(4096x1024) @ W(1024x1024).
// One wave computes a 16x64 tile; K swept in chunks of 4 via fp32 WMMA.
// ---------------------------------------------------------------------------
__global__ void __launch_bounds__(32) proj_kernel(const float* __restrict__ x,
                                                  const float* __restrict__ w,
                                                  float* __restrict__ y) {
  const int lane  = threadIdx.x;
  const int ntile = blockIdx.x & 15;   // 16 tiles of 64 cols
  const int mtile = blockIdx.x >> 4;   // 256 tiles of 16 rows
  const int lm    = lane & 15;
  const int kofs  = (lane >> 4) << 1;
  const int mhi   = (lane >> 4) << 3;
  const int mbase = mtile << 4;
  const int nbase = ntile << 6;

  v8f a0 = {}, a1 = {}, a2 = {}, a3 = {};
  const float* Xp = x + (size_t)(mbase + lm) * DMODEL;
  for (int kc = 0; kc < DMODEL / 4; ++kc) {
    v2f af;
    af[0] = Xp[4 * kc + kofs + 0];
    af[1] = Xp[4 * kc + kofs + 1];
    const float* W0 = w + (size_t)(4 * kc + kofs + 0) * DMODEL + nbase;
    const float* W1 = w + (size_t)(4 * kc + kofs + 1) * DMODEL + nbase;
    { v2f bf; bf[0] = W0[ 0 + lm]; bf[1] = W1[ 0 + lm]; a0 = wmma_f32(af, bf, a0); }
    { v2f bf; bf[0] = W0[16 + lm]; bf[1] = W1[16 + lm]; a1 = wmma_f32(af, bf, a1); }
    { v2f bf; bf[0] = W0[32 + lm]; bf[1] = W1[32 + lm]; a2 = wmma_f32(af, bf, a2); }
    { v2f bf; bf[0] = W0[48 + lm]; bf[1] = W1[48 + lm]; a3 = wmma_f32(af, bf, a3); }
  }
#pragma unroll
  for (int r = 0; r < 8; ++r) {
    int m = r + mhi;
    size_t row = (size_t)(mbase + m) * DMODEL + nbase;
    y[row +  0 + lm] = a0[r];
    y[row + 16 + lm] = a1[r];
    y[row + 32 + lm] = a2[r];
    y[row + 48 + lm] = a3[r];
  }
}

// ---------------------------------------------------------------------------
extern "C" void kernel_launch(void* const* d_in, const int* in_sizes, int n_in,
                              void* d_out, int out_size, void* d_ws, size_t ws_size,
                              hipStream_t stream) {
  const float* q = (const float*)d_in[0];  // (B,T,1024)
  const float* k = (const float*)d_in[1];  // (B,T,256)
  const float* v = (const float*)d_in[2];  // (B,T,256)
  const float* w = (const float*)d_in[3];  // (1024,1024)
  float* out = (float*)d_out;              // (B,T,1024)

  float* qr = (float*)d_ws;                                        // 4M floats
  float* kr = qr + (size_t)BATCH * NHEADS * TSEQ * HEADD;          // 1M floats
  float* xo = kr + (size_t)BATCH * NKV * TSEQ * HEADD;             // 4M floats

  {
    int total = BATCH * TSEQ * NHEADS * HEADD;
    rope_kernel<<<(total + 255) / 256, 256, 0, stream>>>(q, qr, NHEADS);
  }
  {
    int total = BATCH * TSEQ * NKV * HEADD;
    rope_kernel<<<(total + 255) / 256, 256, 0, stream>>>(k, kr, NKV);
  }
  attn_kernel<<<BATCH * NHEADS * (TSEQ / 16), 32, 0, stream>>>(qr, kr, v, xo);
  proj_kernel<<<(BATCH * TSEQ / 16) * (DMODEL / 64), 32, 0, stream>>>(xo, w, out);
}